// GCNLayer_5007931867338
// MI455X (gfx1250) — compile-verified
//
#include <hip/hip_runtime.h>
#include <cstdint>
#include <cstddef>

#define DIM 128

typedef __attribute__((ext_vector_type(2))) float v2f;
typedef __attribute__((ext_vector_type(8))) float v8f;

// ---------------- zero-init workspace ----------------
__global__ void zero_f32_kernel(float* __restrict__ p, size_t n) {
  size_t i = (size_t)blockIdx.x * blockDim.x + threadIdx.x;
  if (i < n) p[i] = 0.0f;
}

// ---------------- in-degree via atomic add ----------------
__global__ void gcn_degree_kernel(const long long* __restrict__ dst,
                                  float* __restrict__ deg, int E) {
  int e = blockIdx.x * blockDim.x + threadIdx.x;
  if (e < E) {
    atomicAdd(&deg[(int)dst[e]], 1.0f);
  }
}

// ---------------- dinv = (deg + 1)^-1/2 ----------------
__global__ void gcn_dinv_kernel(const float* __restrict__ deg,
                                float* __restrict__ dinv, int N) {
  int i = blockIdx.x * blockDim.x + threadIdx.x;
  if (i < N) {
    dinv[i] = rsqrtf(deg[i] + 1.0f);
  }
}

// ---------------- edge scatter: agg[dst] += x[src] * dinv[src] ----------------
// One wave (32 lanes) per edge; each lane moves 4 consecutive floats (float4
// gather, 4 f32 atomics). Gathers + atomics are L2-resident (x fits in 192MB L2).
__global__ __launch_bounds__(256)
void gcn_scatter_kernel(const float* __restrict__ x,
                        const long long* __restrict__ src,
                        const long long* __restrict__ dst,
                        const float* __restrict__ dinv,
                        float* __restrict__ agg, int E) {
  int gid  = blockIdx.x * blockDim.x + threadIdx.x;
  int e    = gid >> 5;
  int lane = gid & 31;
  if (e >= E) return;
  int s = (int)src[e];
  int d = (int)dst[e];
  float sc = dinv[s];
  const float4 xv = *(const float4*)(x + (size_t)s * DIM + lane * 4);
  float* ap = agg + (size_t)d * DIM + lane * 4;
  atomicAdd(ap + 0, xv.x * sc);
  atomicAdd(ap + 1, xv.y * sc);
  atomicAdd(ap + 2, xv.z * sc);
  atomicAdd(ap + 3, xv.w * sc);
}

// ---------------- fused epilogue + GEMM + ReLU on WMMA ----------------
// out[n, :] = relu( ((agg[n,:] + x[n,:]) * dinv[n]) @ W.T )
// 8 waves/block, each wave owns a 16x128 output strip (8 tiles of 16x16 f32
// accumulators), accumulated with V_WMMA_F32_16X16X4_F32 over 32 K-steps.
//
// A-frag layout (16x4 f32, ISA 7.12.2): lane L -> M = L%16,
//   VGPR0 = A[M][k0 + 2*(L/16)], VGPR1 = A[M][k0 + 2*(L/16) + 1]
// B-frag (4x16, mirrored): lane L -> N = L%16,
//   VGPR0 = B[k0 + 2*(L/16)][N] = W[N][k0 + 2*(L/16)]  (contiguous float2 in W)
// C/D layout: VGPR v, lane L -> M = v + 8*(L/16), N = L%16.
__global__ __launch_bounds__(256)
void gcn_gemm_wmma_kernel(const float* __restrict__ x,
                          const float* __restrict__ agg,
                          const float* __restrict__ dinv,
                          const float* __restrict__ W,
                          float* __restrict__ out, int N) {
  const int wave = threadIdx.x >> 5;   // 0..7
  const int lane = threadIdx.x & 31;
  const int half = lane >> 4;          // 0 or 1
  const int l15  = lane & 15;
  const int rowBase = blockIdx.x * 128 + wave * 16;

  const int row  = rowBase + l15;
  const int rowc = (row < N) ? row : (N - 1);   // clamp: keep EXEC uniform for WMMA
  const float scale = dinv[rowc];
  const int kk = half * 2;

  v8f acc[8];
#pragma unroll
  for (int t = 0; t < 8; ++t) acc[t] = (v8f){0.f, 0.f, 0.f, 0.f, 0.f, 0.f, 0.f, 0.f};

  const float* xr = x   + (size_t)rowc * DIM;
  const float* gr = agg + (size_t)rowc * DIM;

#pragma unroll 4
  for (int ks = 0; ks < DIM / 4; ++ks) {
    const int k0 = ks * 4 + kk;
    // A fragment: fused (agg + x) * dinv
    const float2 xa = *(const float2*)(xr + k0);
    const float2 ga = *(const float2*)(gr + k0);
    v2f a;
    a.x = (ga.x + xa.x) * scale;
    a.y = (ga.y + xa.y) * scale;
#pragma unroll
    for (int t = 0; t < 8; ++t) {
      const int n = t * 16 + l15;
      const float2 wb = *(const float2*)(W + (size_t)n * DIM + k0);
      v2f b;
      b.x = wb.x;
      b.y = wb.y;
      acc[t] = __builtin_amdgcn_wmma_f32_16x16x4_f32(
          /*neg_a=*/false, a, /*neg_b=*/false, b,
          /*c_mod=*/(short)0, acc[t], /*reuse_a=*/false, /*reuse_b=*/false);
    }
  }

  // ReLU + store (guarded; WMMA work is done, divergence OK here)
#pragma unroll
  for (int t = 0; t < 8; ++t) {
    const int n = t * 16 + l15;
#pragma unroll
    for (int v = 0; v < 8; ++v) {
      const int m = rowBase + v + half * 8;
      if (m < N) {
        float val = acc[t][v];
        out[(size_t)m * DIM + n] = val > 0.0f ? val : 0.0f;
      }
    }
  }
}

extern "C" void kernel_launch(void* const* d_in, const int* in_sizes, int n_in,
                              void* d_out, int out_size, void* d_ws, size_t ws_size,
                              hipStream_t stream) {
  const float*     x  = (const float*)d_in[0];
  const long long* ei = (const long long*)d_in[1];   // int64 edge_index [2, E]
  const float*     W  = (const float*)d_in[3];

  const int N = in_sizes[0] / DIM;
  const int E = in_sizes[1] / 2;

  float* out  = (float*)d_out;
  float* deg  = (float*)d_ws;        // N floats
  float* dinv = deg + N;             // N floats
  float* agg  = dinv + N;            // N*DIM floats

  const long long* src = ei;
  const long long* dst = ei + E;

  // zero deg + dinv + agg (contiguous)
  const size_t nz = (size_t)N * (DIM + 2);
  zero_f32_kernel<<<(unsigned)((nz + 1023) / 1024), 1024, 0, stream>>>((float*)d_ws, nz);

  gcn_degree_kernel<<<(E + 255) / 256, 256, 0, stream>>>(dst, deg, E);
  gcn_dinv_kernel<<<(N + 255) / 256, 256, 0, stream>>>(deg, dinv, N);

  const long long scatterThreads = (long long)E * 32;
  gcn_scatter_kernel<<<(unsigned)((scatterThreads + 255) / 256), 256, 0, stream>>>(
      x, src, dst, dinv, agg, E);

  gcn_gemm_wmma_kernel<<<(N + 127) / 128, 256, 0, stream>>>(x, agg, dinv, W, out, N);
}